// LocallyConnected3D_27333171872132
// MI455X (gfx1250) — compile-verified
//
#include <hip/hip_runtime.h>

// LocallyConnected3D (B=4, D=H=W=32, Cin=8, 3x3x3 valid, F=16)
// out[b,l,f] = sum_p patches[b,l,p] * kernel[l,p,f] + bias[l,f]
// L=27000 locations, P=216, F=16.
//
// Memory bound: 373MB of per-location weights streamed once -> ~16.5us floor
// at 23.3 TB/s. Compute mapped to V_WMMA_F32_16X16X4_F32 (exact f32 math),
// one wave32 per output location, 54 fully-unrolled K-steps of 4 so every
// load is an immediate-offset global_load (no per-step address math).

typedef __attribute__((ext_vector_type(2))) float v2f;
typedef __attribute__((ext_vector_type(8))) float v8f;

#define LLOC 27000
#define PP   216
#define FF   16

__global__ __launch_bounds__(256)
void lc3d_wmma_kernel(const float* __restrict__ x,
                      const float* __restrict__ w,
                      const float* __restrict__ bias,
                      float* __restrict__ out) {
  const int lane = threadIdx.x & 31;
  const int wave = threadIdx.x >> 5;
  const int l    = blockIdx.x * 8 + wave;   // grid is exact: 3375*8 = 27000

  // decode spatial location (od,oh,ow), each in [0,30)
  const int ow = l % 30;
  const int oh = (l / 30) % 30;
  const int od = l / 900;

  // ---- WMMA f32 16x16x4 operand layouts (wave32) ----
  // A (16x4, MxK): lane holds M = lane&15; lanes 0-15 carry K=0,1,
  //                lanes 16-31 carry K=2,3.
  // B (4x16, KxN): lane holds N = lane&15; VGPR r: lanes 0-15 = row K=r,
  //                lanes 16-31 = row K=r+2.
  // C/D (16x16):   element r of v8f: lanes 0-15 = M=r, lanes 16-31 = M=r+8.
  const int row  = lane & 15;     // A-row; batch index when < 4
  const int col  = lane & 15;     // B-column / output feature f
  const int half = lane >> 4;     // 0 -> K offset 0,1 ; 1 -> K offset 2,3

  // Rows 4..15 of A only feed C rows 4..15 which are never stored, so no
  // masking is needed -- just clamp to a valid address (finite data, no NaNs).
  const int b = row & 3;

  // x strides (B,D,H,W,Cin)=(4,32,32,32,8): b:262144, d:8192, h:256, w:8, c:1
  const float* xb = x + ((((size_t)b * 32 + od) * 32 + oh) * 32 + ow) * 8;
  // p0 = 4t + 2*half shares the (kd,kh,kw) tap with p0 = 4t, so the tap
  // offset is wave-uniform and the per-lane part is just +2*half channels:
  const float* xh = xb + 2 * half;

  const float* wb   = w + (size_t)l * (PP * FF);  // this location's 216x16 block
  const float* wcol = wb + 32 * half + col;       // per-lane B base

  // bias for this lane's output column; issued early, consumed after the loop
  const float bval = bias[(size_t)l * FF + col];

  v8f acc = {0.f, 0.f, 0.f, 0.f, 0.f, 0.f, 0.f, 0.f};

#pragma unroll
  for (int t = 0; t < PP / 4; ++t) {
    // ---- compile-time constant offsets (full unroll) ----
    const int p0  = 4 * t;            // uniform K base of this step
    const int tap = p0 >> 3;          // kd*9 + kh*3 + kw  (wave-uniform)
    const int c0  = p0 & 7;           // 0 or 4
    const int kd  = tap / 9;
    const int r9  = tap - kd * 9;
    const int kh  = r9 / 3;
    const int kw  = r9 - kh * 3;
    const int axoff = kd * 8192 + kh * 256 + kw * 8 + c0;

    // A operand: one aligned 8-byte load (x is L2-resident: 4.2MB vs 192MB L2)
    const v2f a = *(const v2f*)(xh + axoff);

    // B operand: weight rows p0(+2*half), p0+1(+2*half), column `col`.
    // Streamed exactly once -> nontemporal.
    v2f bv;
    bv.x = __builtin_nontemporal_load(wcol + 64 * t);
    bv.y = __builtin_nontemporal_load(wcol + 64 * t + 16);

    // D = A*B + C   (v_wmma_f32_16x16x4_f32)
    acc = __builtin_amdgcn_wmma_f32_16x16x4_f32(
        /*neg_a=*/false, a, /*neg_b=*/false, bv,
        /*c_mod=*/(short)0, acc, /*reuse_a=*/false, /*reuse_b=*/false);
  }

  // C rows M=0..3 live in acc[0..3] on lanes 0-15 (N = f = col).
  if (lane < 16) {
#pragma unroll
    for (int bb = 0; bb < 4; ++bb) {
      __builtin_nontemporal_store(acc[bb] + bval,
          out + (((size_t)bb * LLOC + l) * FF + col));
    }
  }
}

extern "C" void kernel_launch(void* const* d_in, const int* in_sizes, int n_in,
                              void* d_out, int out_size, void* d_ws, size_t ws_size,
                              hipStream_t stream) {
  const float* x    = (const float*)d_in[0];
  const float* w    = (const float*)d_in[1];
  const float* bias = (const float*)d_in[2];
  float* out        = (float*)d_out;

  dim3 grid(LLOC / 8);   // 3375 blocks x 8 wave32 = 27000 locations (exact)
  dim3 block(256);
  lc3d_wmma_kernel<<<grid, block, 0, stream>>>(x, w, bias, out);
}